// GIN_60868276519674
// MI455X (gfx1250) — compile-verified
//
#include <hip/hip_runtime.h>
#include <hip/hip_bf16.h>

// ---------------------------------------------------------------------------
// GIN (2-layer) + projection for MI455X (gfx1250, wave32).
//  - Sparse aggregation: per-edge wave, float4 gather + global_atomic_add_f32
//    (feature matrix is 51.2 MB -> L2-resident on 192 MB L2).
//  - Dense GEMMs: fp32 WMMA (v_wmma_f32_16x16x4_f32). Templated on N so all
//    B/C addressing folds to immediate offsets; 5 row-tiles per block reuse
//    the per-wave B register fragments (160 WMMAs / wave).
// ---------------------------------------------------------------------------

typedef __attribute__((ext_vector_type(2))) float v2f;
typedef __attribute__((ext_vector_type(8))) float v8f;

#define HID 128
#define OUTD 64

// x[node][0:128] = embed[ids[node]][0:128]; one wave per node, float4 lanes.
__global__ void embed_lookup_kernel(const int* __restrict__ ids,
                                    const float* __restrict__ embed,
                                    float* __restrict__ x, int nNodes) {
  int t    = blockIdx.x * blockDim.x + threadIdx.x;
  int node = t >> 5;
  int lane = t & 31;
  if (node >= nNodes) return;
  int id = ids[node];
  const float4* src = (const float4*)(embed + (size_t)id * HID);
  float4* dst       = (float4*)(x + (size_t)node * HID);
  dst[lane] = src[lane];
}

// agg = (1 + eps) * x   (doubles as the zero-init of the accumulator)
__global__ void self_scale_kernel(const float* __restrict__ x,
                                  const float* __restrict__ eps,
                                  float* __restrict__ agg, int n4) {
  int i = blockIdx.x * blockDim.x + threadIdx.x;
  if (i >= n4) return;
  float s  = 1.0f + eps[0];
  float4 v = ((const float4*)x)[i];
  v.x *= s; v.y *= s; v.z *= s; v.w *= s;
  ((float4*)agg)[i] = v;
}

// agg[dst[e]][:] += x[src[e]][:]; one wave per edge, lane handles 4 floats.
__global__ void edge_scatter_kernel(const float* __restrict__ x,
                                    const int* __restrict__ esrc,
                                    const int* __restrict__ edst,
                                    float* __restrict__ agg, int nEdges) {
  int t    = blockIdx.x * blockDim.x + threadIdx.x;
  int edge = t >> 5;
  int lane = t & 31;
  if (edge >= nEdges) return;
  int s = esrc[edge];
  int d = edst[edge];
  float4 v = ((const float4*)(x + (size_t)s * HID))[lane];
  float* drow = agg + (size_t)d * HID + lane * 4;
  atomicAdd(drow + 0, v.x);
  atomicAdd(drow + 1, v.y);
  atomicAdd(drow + 2, v.z);
  atomicAdd(drow + 3, v.w);
}

// C = A(rows x 128) * B(128 x N) + bias (+ReLU), fp32 WMMA.
// Block: RPB 16-row tiles; wave w covers columns [16w, 16w+16).
// blockDim.x = 32 * (N/16).
//
// fp32 WMMA 16x16x4 fragment layout (ISA 7.12.2):
//   A 16x4 : lanes 0-15 -> M=lane, K={k,k+1}; lanes 16-31 -> M=lane-16, K={k+2,k+3}
//   B 4x16 : lanes 0-15 -> N=lane, K={k,k+1}; lanes 16-31 -> N=lane-16, K={k+2,k+3}
//   C 16x16: VGPR r: lanes 0-15 -> (M=r, N=lane); lanes 16-31 -> (M=r+8, N=lane-16)
#define LDS_STRIDE 132  // 128+4 pad: lane l hits bank (4*l + k) % 64 -> conflict-free

template <int N, int RPB, bool RELU>
__global__ void gemm_wmma_f32_kernel(const float* __restrict__ A,
                                     const float* __restrict__ B,
                                     const float* __restrict__ bias,
                                     float* __restrict__ C,
                                     int nRows) {
  __shared__ float As[RPB * 16 * LDS_STRIDE];

  const int tid  = threadIdx.x;
  const int lane = tid & 31;
  const int wave = tid >> 5;
  const int hi   = lane >> 4;   // wave half: selects K pair {2hi, 2hi+1}
  const int l15  = lane & 15;
  const size_t rowBase = (size_t)blockIdx.x * (RPB * 16);

  // Stage RPB x 16 x 128 A tiles into LDS (padded stride).
  for (int i = tid; i < RPB * 16 * HID; i += blockDim.x) {
    int r = i >> 7;            // row within block's tiles
    int c = i & (HID - 1);
    size_t row = rowBase + r;
    As[r * LDS_STRIDE + c] = (row < (size_t)nRows) ? A[row * HID + c] : 0.0f;
  }
  __syncthreads();

  const int col = wave * 16 + l15;

  // Preload B fragments once per wave: bfrag[k4] = {B[4k4+2hi][col], B[4k4+2hi+1][col]}.
  // N is compile-time -> these become global_load_b32 with immediate offsets.
  const float* Bp = B + (size_t)(hi * 2) * N + col;
  v2f bfrag[HID / 4];
#pragma unroll
  for (int k4 = 0; k4 < HID / 4; ++k4) {
    bfrag[k4][0] = Bp[k4 * 4 * N];
    bfrag[k4][1] = Bp[k4 * 4 * N + N];
  }
  const float bv = bias[col];
  const float* Ap = As + l15 * LDS_STRIDE + hi * 2;

#pragma unroll
  for (int r = 0; r < RPB; ++r) {
    const size_t tileRow = rowBase + (size_t)r * 16;
    if (tileRow < (size_t)nRows) {        // uniform per wave; full EXEC inside
      v8f acc = {};
#pragma unroll
      for (int k4 = 0; k4 < HID / 4; ++k4) {
        v2f a = *(const v2f*)(Ap + r * 16 * LDS_STRIDE + k4 * 4);  // ds_load_b64
        acc = __builtin_amdgcn_wmma_f32_16x16x4_f32(
            false, a, false, bfrag[k4], (short)0, acc, false, false);
      }
      float* Cp = C + (tileRow + hi * 8) * (size_t)N + col;
#pragma unroll
      for (int rr = 0; rr < 8; ++rr) {
        float v = acc[rr] + bv;
        if (RELU) v = fmaxf(v, 0.0f);
        Cp[(size_t)rr * N] = v;           // immediate offsets (N compile-time)
      }
    }
  }
}

#define ROWS_PER_BLOCK 5   // 100000 rows = 1250 blocks * 5 tiles * 16 rows exactly

extern "C" void kernel_launch(void* const* d_in, const int* in_sizes, int n_in,
                              void* d_out, int out_size, void* d_ws, size_t ws_size,
                              hipStream_t stream) {
  const int*   x_ids = (const int*)  d_in[0];
  const int*   esrc  = (const int*)  d_in[1];
  const int*   edst  = (const int*)  d_in[2];
  const float* embed = (const float*)d_in[3];
  const float* eps0  = (const float*)d_in[4];
  const float* w1_0  = (const float*)d_in[5];
  const float* b1_0  = (const float*)d_in[6];
  const float* w2_0  = (const float*)d_in[7];
  const float* b2_0  = (const float*)d_in[8];
  const float* eps1  = (const float*)d_in[9];
  const float* w1_1  = (const float*)d_in[10];
  const float* b1_1  = (const float*)d_in[11];
  const float* w2_1  = (const float*)d_in[12];
  const float* b2_1  = (const float*)d_in[13];
  const float* wp    = (const float*)d_in[14];
  const float* bp    = (const float*)d_in[15];

  const int nNodes = in_sizes[0];
  const int nEdges = in_sizes[1];

  float* out  = (float*)d_out;
  float* yhat = out;                              // [nNodes, 64]
  float* xOut = out + (size_t)nNodes * OUTD;      // [nNodes, 128] (final x)

  float* bufX   = (float*)d_ws;                   // current features
  float* bufAgg = bufX   + (size_t)nNodes * HID;  // aggregation
  float* bufT   = bufAgg + (size_t)nNodes * HID;  // MLP hidden

  const int n4          = nNodes * (HID / 4);     // float4 count
  const int rowTiles    = (nNodes + 15) / 16;
  const int gemmBlocks  = (rowTiles + ROWS_PER_BLOCK - 1) / ROWS_PER_BLOCK;
  const int embedBlocks = (nNodes * 32 + 255) / 256;
  const int scaleBlocks = (n4 + 255) / 256;
  const int edgeBlocks  = (int)(((size_t)nEdges * 32 + 255) / 256);

  // 1) embedding lookup
  embed_lookup_kernel<<<embedBlocks, 256, 0, stream>>>(x_ids, embed, bufX, nNodes);

  // 2) GIN layer 0
  self_scale_kernel<<<scaleBlocks, 256, 0, stream>>>(bufX, eps0, bufAgg, n4);
  edge_scatter_kernel<<<edgeBlocks, 256, 0, stream>>>(bufX, esrc, edst, bufAgg, nEdges);
  gemm_wmma_f32_kernel<HID, ROWS_PER_BLOCK, false>
      <<<gemmBlocks, 32 * (HID / 16), 0, stream>>>(bufAgg, w1_0, b1_0, bufT, nNodes);
  gemm_wmma_f32_kernel<HID, ROWS_PER_BLOCK, true>
      <<<gemmBlocks, 32 * (HID / 16), 0, stream>>>(bufT, w2_0, b2_0, bufX, nNodes);

  // 3) GIN layer 1 (final features straight into d_out's x region)
  self_scale_kernel<<<scaleBlocks, 256, 0, stream>>>(bufX, eps1, bufAgg, n4);
  edge_scatter_kernel<<<edgeBlocks, 256, 0, stream>>>(bufX, esrc, edst, bufAgg, nEdges);
  gemm_wmma_f32_kernel<HID, ROWS_PER_BLOCK, false>
      <<<gemmBlocks, 32 * (HID / 16), 0, stream>>>(bufAgg, w1_1, b1_1, bufT, nNodes);
  gemm_wmma_f32_kernel<HID, ROWS_PER_BLOCK, true>
      <<<gemmBlocks, 32 * (HID / 16), 0, stream>>>(bufT, w2_1, b2_1, xOut, nNodes);

  // 4) projection: y_hat = x @ wp + bp  (N = 64 -> 4 waves per block)
  gemm_wmma_f32_kernel<OUTD, ROWS_PER_BLOCK, false>
      <<<gemmBlocks, 32 * (OUTD / 16), 0, stream>>>(xOut, wp, bp, yhat, nNodes);
}